// SuperGATEncoder_36395552866371
// MI455X (gfx1250) — compile-verified
//
#include <hip/hip_runtime.h>
#include <hip/hip_bf16.h>

// ---------------------------------------------------------------------------
// SuperGAT encoder for MI455X (gfx1250, wave32).
//   GEMMs:      V_WMMA_F32_16X16X4_F32, B fragments register-resident,
//               A fragments staged per row tile (batched load clause ->
//               overlapped 32-deep WMMA chain), 5 row tiles per wave.
//   Edge phase: wave-per-edge gather + ordered-uint atomic max + f32 atomic add
// ---------------------------------------------------------------------------

#define N_NODES 50000
#define N_EDGES 800000
#define E_TOT   (N_EDGES + N_NODES)   // self loops appended
#define IN_CH   128
#define HID     64                    // per-head channels, layer 1
#define HEADS   2
#define HC1     (HEADS * HID)         // 128
#define OUT_CH  32
#define MROWS   5                     // row tiles per wave (3125 % 5 == 0)

typedef float v2f __attribute__((ext_vector_type(2)));
typedef float v8f __attribute__((ext_vector_type(8)));

// Ordered-uint encoding so atomicMax(uint) == float max (handles negatives).
__device__ __forceinline__ unsigned encf(float f) {
    unsigned u = __float_as_uint(f);
    return (u & 0x80000000u) ? ~u : (u | 0x80000000u);
}
__device__ __forceinline__ float decf(unsigned u) {
    unsigned b = (u & 0x80000000u) ? (u & 0x7FFFFFFFu) : ~u;
    return __uint_as_float(b);
}
#define ENC_NEG_INF 0x007FFFFFu   // encf(-inf)

// ---------------------------------------------------------------------------
// WMMA f32 GEMM: Y[M,N] = X[M,K] @ W[K,N].  K compile-time (128 here).
// One wave owns one 16-wide column stripe and MROWS consecutive 16x16 tiles.
//   - B (K x 16) fragments loaded once into registers, reused across tiles.
//   - A fragments for a whole row tile staged into a register array first,
//     so loads issue as one clause and the WMMA chain drains loadcnt
//     incrementally instead of stalling per step.
//
// Fragment layouts (wave32):
//   A 16x4 : lanes 0-15 M=0..15 (V0:K=0,V1:K=1); lanes 16-31 (V0:K=2,V1:K=3)
//   B 4x16 : V0: lanes0-15 K=0 / lanes16-31 K=2; V1: K=1 / K=3; N = lane%16
//   D 16x16: VGPR r: lanes0-15 M=r, lanes16-31 M=r+8; N = lane%16
// ---------------------------------------------------------------------------
template<int K>
__global__ void wmma_gemm_k(const float* __restrict__ X,
                            const float* __restrict__ W,
                            float* __restrict__ Y,
                            int M, int N)
{
    constexpr int KSTEPS = K / 4;
    int lane = threadIdx.x & 31;
    int wid  = blockIdx.x * (blockDim.x >> 5) + (threadIdx.x >> 5);
    int colTiles  = N >> 4;
    int rowGroups = (M >> 4) / MROWS;          // exact for M = 50000
    if (wid >= colTiles * rowGroups) return;   // wave-uniform

    int ct = wid % colTiles;
    int rg = wid / colTiles;
    int colBase = ct << 4;
    int m16  = lane & 15;
    int koff = (lane < 16) ? 0 : 2;
    int rowOff = (lane < 16) ? 0 : 8;

    // ---- load B fragments once, keep in registers ----
    v2f bf[KSTEPS];
    const float* wp = W + (long long)koff * N + colBase + m16;
#pragma unroll
    for (int kk = 0; kk < KSTEPS; ++kk) {
        bf[kk].x = wp[(long long)(4 * kk) * N];
        bf[kk].y = wp[(long long)(4 * kk + 1) * N];
    }

    // ---- sweep MROWS row tiles, reusing B ----
#pragma unroll
    for (int r = 0; r < MROWS; ++r) {
        int rowBase = (rg * MROWS + r) << 4;
        const float* xrow = X + (long long)(rowBase + m16) * K + koff;

        // Stage all A fragments: independent dest VGPRs -> one load clause.
        v2f af[KSTEPS];
#pragma unroll
        for (int kk = 0; kk < KSTEPS; ++kk) {
            af[kk].x = xrow[4 * kk];
            af[kk].y = xrow[4 * kk + 1];
        }

        v8f c = {};
#pragma unroll
        for (int kk = 0; kk < KSTEPS; ++kk)
            c = __builtin_amdgcn_wmma_f32_16x16x4_f32(
                    false, af[kk], false, bf[kk], (short)0, c, false, false);

#pragma unroll
        for (int q = 0; q < 8; ++q)
            Y[(long long)(rowBase + q + rowOff) * N + colBase + m16] = c[q];
    }
}

// ---------------------------------------------------------------------------
// Init helpers (deterministic re-init of workspace every call)
// ---------------------------------------------------------------------------
__global__ void init_u32_k(unsigned* p, unsigned v, int n) {
    int i = blockIdx.x * blockDim.x + threadIdx.x;
    if (i < n) p[i] = v;
}
__global__ void init_f32_k(float* p, float v, int n) {
    int i = blockIdx.x * blockDim.x + threadIdx.x;
    if (i < n) p[i] = v;
}

// ---------------------------------------------------------------------------
// Edge pass 1: alpha[e,h] = leaky(0.2)( (aL.xj + aR.xi) * sigmoid(xi.xj) )
// and running segment-max into amax_enc[dst,h].
// One wave per edge; lane owns H*C/32 contiguous channels.
// ---------------------------------------------------------------------------
template<int H, int C>
__global__ void edge_alpha_k(const float* __restrict__ xh,
                             const int*   __restrict__ ei,
                             const float* __restrict__ attL,
                             const float* __restrict__ attR,
                             float*    __restrict__ alpha,
                             unsigned* __restrict__ amax)
{
    constexpr int HC  = H * C;
    constexpr int VEC = HC / 32;   // channels per lane
    constexpr int LPH = 32 / H;    // lanes per head
    int lane = threadIdx.x & 31;
    int e    = blockIdx.x * (blockDim.x >> 5) + (threadIdx.x >> 5);
    if (e >= E_TOT) return;

    int s, d;
    if (e < N_EDGES) { s = ei[e]; d = ei[N_EDGES + e]; }
    else             { s = e - N_EDGES; d = s; }

    const float* xj = xh + (long long)s * HC + lane * VEC;
    const float* xi = xh + (long long)d * HC + lane * VEC;
    float pl = 0.f, pa = 0.f;
#pragma unroll
    for (int v = 0; v < VEC; ++v) {
        float aj = xj[v], ai = xi[v];
        int   ch = lane * VEC + v;         // == h*C + c (contiguous layout)
        pl += ai * aj;
        pa += attL[ch] * aj + attR[ch] * ai;
    }
#pragma unroll
    for (int m = LPH >> 1; m >= 1; m >>= 1) {
        pl += __shfl_xor(pl, m, 32);
        pa += __shfl_xor(pa, m, 32);
    }
    if ((lane & (LPH - 1)) == 0) {
        int   h = lane / LPH;
        float a = pa * (1.f / (1.f + __expf(-pl)));   // MX gating
        a = (a > 0.f) ? a : 0.2f * a;                 // leaky relu
        alpha[(long long)e * H + h] = a;
        atomicMax(&amax[(long long)d * H + h], encf(a));
    }
}

// ---------------------------------------------------------------------------
// Edge pass 2: w = exp(alpha - amax[dst]); denom[dst,h] += w;
//              accum[dst,h,c] += w * xj[h,c]
// ---------------------------------------------------------------------------
template<int H, int C>
__global__ void edge_accum_k(const float* __restrict__ xh,
                             const int*   __restrict__ ei,
                             const float* __restrict__ alpha,
                             const unsigned* __restrict__ amax,
                             float* __restrict__ denom,
                             float* __restrict__ accum)
{
    constexpr int HC  = H * C;
    constexpr int VEC = HC / 32;
    constexpr int LPH = 32 / H;
    int lane = threadIdx.x & 31;
    int e    = blockIdx.x * (blockDim.x >> 5) + (threadIdx.x >> 5);
    if (e >= E_TOT) return;

    int s, d;
    if (e < N_EDGES) { s = ei[e]; d = ei[N_EDGES + e]; }
    else             { s = e - N_EDGES; d = s; }

    int   h  = lane / LPH;
    float a  = alpha[(long long)e * H + h];
    float mx = decf(amax[(long long)d * H + h]);
    float w  = __expf(a - mx);
    if ((lane & (LPH - 1)) == 0)
        atomicAdd(&denom[(long long)d * H + h], w);

    const float* xj  = xh    + (long long)s * HC + lane * VEC;
    float*       acc = accum + (long long)d * HC + lane * VEC;
#pragma unroll
    for (int v = 0; v < VEC; ++v)
        atomicAdd(&acc[v], w * xj[v]);
}

// ---------------------------------------------------------------------------
// Finalize: out = accum / (denom + 1e-16) + bias  (optional ELU)
// ---------------------------------------------------------------------------
__global__ void finalize_k(const float* __restrict__ accum,
                           const float* __restrict__ denom,
                           const float* __restrict__ bias,
                           float* __restrict__ out,
                           int n, int H, int C, int applyElu)
{
    int HC  = H * C;
    int idx = blockIdx.x * blockDim.x + threadIdx.x;
    if (idx >= n * HC) return;
    int node = idx / HC;
    int ch   = idx - node * HC;
    int h    = ch / C;
    float v = accum[idx] / (denom[node * H + h] + 1e-16f) + bias[ch];
    if (applyElu) v = (v > 0.f) ? v : (__expf(v) - 1.f);
    out[idx] = v;
}

// ---------------------------------------------------------------------------
extern "C" void kernel_launch(void* const* d_in, const int* in_sizes, int n_in,
                              void* d_out, int out_size, void* d_ws, size_t ws_size,
                              hipStream_t stream) {
    const float* x      = (const float*)d_in[0];
    const int*   ei     = (const int*)  d_in[1];
    const float* W1     = (const float*)d_in[2];
    const float* attl1  = (const float*)d_in[3];
    const float* attr1  = (const float*)d_in[4];
    const float* b1     = (const float*)d_in[5];
    const float* Wmu    = (const float*)d_in[6];
    const float* attlmu = (const float*)d_in[7];
    const float* attrmu = (const float*)d_in[8];
    const float* bmu    = (const float*)d_in[9];
    const float* Wls    = (const float*)d_in[10];
    const float* attlls = (const float*)d_in[11];
    const float* attrls = (const float*)d_in[12];
    const float* bls    = (const float*)d_in[13];
    float* out = (float*)d_out;                     // [mu | logstd]

    // ---- workspace partition (floats) ----
    float* ws = (float*)d_ws;
    float*    xh1    = ws;                                  // 6,400,000  (reused for xh_mu/xh_ls)
    float*    accum1 = ws + 6400000;                        // 6,400,000  (becomes h in-place)
    float*    alpha  = ws + 12800000;                       // 1,700,000
    unsigned* amax   = (unsigned*)(ws + 14500000);          //   100,000
    float*    denom  = ws + 14600000;                       //   100,000
    float*    accum2 = ws + 14700000;                       // 3,200,000  (mu | logstd)

    const int EBLK = (E_TOT + 7) / 8;          // 8 waves (edges) per 256-thr block
    const int T256 = 256;
    const int rowGroups = (N_NODES / 16) / MROWS;           // 625, exact

    // ===== Layer 1: xh1 = x @ W1 (50000x128 @ 128x128) =====
    {
        int waves = rowGroups * (HC1 / 16);                 // 5000
        wmma_gemm_k<IN_CH><<<(waves + 7) / 8, T256, 0, stream>>>(x, W1, xh1, N_NODES, HC1);
    }
    init_u32_k<<<(N_NODES * HEADS + 255) / 256, T256, 0, stream>>>(amax, ENC_NEG_INF, N_NODES * HEADS);
    init_f32_k<<<(N_NODES * HEADS + 255) / 256, T256, 0, stream>>>(denom, 0.f, N_NODES * HEADS);
    init_f32_k<<<(N_NODES * HC1 + 255) / 256, T256, 0, stream>>>(accum1, 0.f, N_NODES * HC1);

    edge_alpha_k<HEADS, HID><<<EBLK, T256, 0, stream>>>(xh1, ei, attl1, attr1, alpha, amax);
    edge_accum_k<HEADS, HID><<<EBLK, T256, 0, stream>>>(xh1, ei, alpha, amax, denom, accum1);
    finalize_k<<<(N_NODES * HC1 + 255) / 256, T256, 0, stream>>>(
        accum1, denom, b1, accum1, N_NODES, HEADS, HID, /*elu=*/1);   // h in-place

    // ===== Layer 2 projections: xh_mu / xh_ls = h @ Wmu / Wls =====
    float* xh_mu = xh1;
    float* xh_ls = xh1 + (size_t)N_NODES * OUT_CH;
    {
        int waves = rowGroups * (OUT_CH / 16);              // 1250
        wmma_gemm_k<HC1><<<(waves + 7) / 8, T256, 0, stream>>>(accum1, Wmu, xh_mu, N_NODES, OUT_CH);
        wmma_gemm_k<HC1><<<(waves + 7) / 8, T256, 0, stream>>>(accum1, Wls, xh_ls, N_NODES, OUT_CH);
    }
    init_u32_k<<<(2 * N_NODES + 255) / 256, T256, 0, stream>>>(amax, ENC_NEG_INF, 2 * N_NODES);
    init_f32_k<<<(2 * N_NODES + 255) / 256, T256, 0, stream>>>(denom, 0.f, 2 * N_NODES);
    init_f32_k<<<(2 * N_NODES * OUT_CH + 255) / 256, T256, 0, stream>>>(accum2, 0.f, 2 * N_NODES * OUT_CH);

    // mu head
    edge_alpha_k<1, OUT_CH><<<EBLK, T256, 0, stream>>>(xh_mu, ei, attlmu, attrmu, alpha, amax);
    edge_accum_k<1, OUT_CH><<<EBLK, T256, 0, stream>>>(xh_mu, ei, alpha, amax, denom, accum2);
    // logstd head
    edge_alpha_k<1, OUT_CH><<<EBLK, T256, 0, stream>>>(xh_ls, ei, attlls, attrls,
                                                       alpha + E_TOT, amax + N_NODES);
    edge_accum_k<1, OUT_CH><<<EBLK, T256, 0, stream>>>(xh_ls, ei, alpha + E_TOT, amax + N_NODES,
                                                       denom + N_NODES,
                                                       accum2 + (size_t)N_NODES * OUT_CH);

    finalize_k<<<(N_NODES * OUT_CH + 255) / 256, T256, 0, stream>>>(
        accum2, denom, bmu, out, N_NODES, 1, OUT_CH, 0);
    finalize_k<<<(N_NODES * OUT_CH + 255) / 256, T256, 0, stream>>>(
        accum2 + (size_t)N_NODES * OUT_CH, denom + N_NODES, bls,
        out + (size_t)N_NODES * OUT_CH, N_NODES, 1, OUT_CH, 0);
}